// OTRouter_41120016892130
// MI455X (gfx1250) — compile-verified
//
#include <hip/hip_runtime.h>
#include <hip/hip_bf16.h>
#include <math.h>

// ---------------------------------------------------------------------------
// OT (Sinkhorn) router for MI455X / gfx1250.
//   x:      (8, 8192, 2048) f32  -> N = 65536 tokens, D = 2048
//   gate_w: (64, 2048) f32       -> E = 64 experts
// Outputs (concatenated flat, f32):
//   dispatch (N*64) | indices-as-float (N*8) | load_loss (1)
// ---------------------------------------------------------------------------

typedef __attribute__((ext_vector_type(2))) float v2f;
typedef __attribute__((ext_vector_type(8))) float v8f;

#define N_TOK 65536L
#define DIMK  2048
#define N_EXP 64
#define N_ACT 8
#define NEG_INV_EPS (-20.0f)   // -1/epsilon, epsilon = 0.05

// ---------------------------------------------------------------------------
// Kernel 1: cost matrix  c[t][e] = -(x[t] . gate_w[e]) / eps   via f32 WMMA.
// One wave => 16 tokens x 64 experts (4 WMMA tiles), K-loop in steps of 4.
// A-frag (16x4 f32): lane l holds M = l&15, K = 2*(l>>4) .. +1  -> b64/lane.
// B-frag (4x16 f32): lane l holds N = l&15, K = 2*(l>>4) .. +1  -> b64/lane.
// C/D (16x16 f32): VGPR j, lane l -> M = (l>>4)*8 + j, N = l&15.
// ---------------------------------------------------------------------------
__global__ __launch_bounds__(256) void ot_gemm_cost(const float* __restrict__ x,
                                                    const float* __restrict__ gw,
                                                    float* __restrict__ cost) {
  const int lane = threadIdx.x & 31;
  const int wave = threadIdx.x >> 5;
  const int m    = lane & 15;
  const int kh   = lane >> 4;                      // 0 or 1
  const long tok0 = ((long)blockIdx.x * 8 + wave) * 16;

  const float* ax = x  + (tok0 + m) * DIMK + 2 * kh;
  const float* b0 = gw + ( 0 + m) * DIMK + 2 * kh;
  const float* b1 = gw + (16 + m) * DIMK + 2 * kh;
  const float* b2 = gw + (32 + m) * DIMK + 2 * kh;
  const float* b3 = gw + (48 + m) * DIMK + 2 * kh;

  v8f acc0 = {}; v8f acc1 = {}; v8f acc2 = {}; v8f acc3 = {};

  for (int k0 = 0; k0 < DIMK; k0 += 4) {
    __builtin_prefetch(ax + k0 + 512, 0, 0);       // global_prefetch_b8, ~2KB ahead
    v2f a  = *(const v2f*)(ax + k0);
    v2f w0 = *(const v2f*)(b0 + k0);
    v2f w1 = *(const v2f*)(b1 + k0);
    v2f w2 = *(const v2f*)(b2 + k0);
    v2f w3 = *(const v2f*)(b3 + k0);
    acc0 = __builtin_amdgcn_wmma_f32_16x16x4_f32(false, a, false, w0, (short)0, acc0, false, false);
    acc1 = __builtin_amdgcn_wmma_f32_16x16x4_f32(false, a, false, w1, (short)0, acc1, false, false);
    acc2 = __builtin_amdgcn_wmma_f32_16x16x4_f32(false, a, false, w2, (short)0, acc2, false, false);
    acc3 = __builtin_amdgcn_wmma_f32_16x16x4_f32(false, a, false, w3, (short)0, acc3, false, false);
  }

#pragma unroll
  for (int i = 0; i < 8; ++i) {
    float* crow = cost + (tok0 + kh * 8 + i) * N_EXP + m;
    crow[0]  = NEG_INV_EPS * acc0[i];
    crow[16] = NEG_INV_EPS * acc1[i];
    crow[32] = NEG_INV_EPS * acc2[i];
    crow[48] = NEG_INV_EPS * acc3[i];
  }
}

// ---------------------------------------------------------------------------
// Kernel 2 (x20): one Sinkhorn iteration, row half + column partial sums.
//   r_t      = LSE_e(c[t][e] - s[e])                       (written to r)
//   colsum_e += sum_t exp(c[t][e] - s[e] - r_t)            (atomic f32)
// One thread per token; 256 threads/block, 256 blocks.
// ---------------------------------------------------------------------------
__global__ __launch_bounds__(256) void ot_sinkhorn_row(const float* __restrict__ cost,
                                                       const float* __restrict__ s,
                                                       float* __restrict__ r,
                                                       float* __restrict__ colsum) {
  __shared__ float s_sh[N_EXP];
  __shared__ float part[N_EXP];
  const int tid = threadIdx.x;
  if (tid < N_EXP) { s_sh[tid] = s[tid]; part[tid] = 0.0f; }
  __syncthreads();

  const long t = (long)blockIdx.x * blockDim.x + tid;
  const float4* crow = (const float4*)(cost + t * N_EXP);

  float v[N_EXP];
  float mx = -INFINITY;
#pragma unroll
  for (int i = 0; i < 16; ++i) {
    float4 q = crow[i];
    v[4*i+0] = q.x - s_sh[4*i+0];
    v[4*i+1] = q.y - s_sh[4*i+1];
    v[4*i+2] = q.z - s_sh[4*i+2];
    v[4*i+3] = q.w - s_sh[4*i+3];
    mx = fmaxf(mx, fmaxf(fmaxf(v[4*i+0], v[4*i+1]), fmaxf(v[4*i+2], v[4*i+3])));
  }
  float sum = 0.0f;
#pragma unroll
  for (int e = 0; e < N_EXP; ++e) { v[e] = expf(v[e] - mx); sum += v[e]; }
  r[t] = mx + logf(sum);
  const float inv = 1.0f / sum;
#pragma unroll
  for (int e = 0; e < N_EXP; ++e) atomicAdd(&part[e], v[e] * inv);  // ds_add_f32
  __syncthreads();
  if (tid < N_EXP) atomicAdd(&colsum[tid], part[tid]);              // global_atomic_add_f32
}

// Column half of the iteration: s_e <- s_e + log(colsum_e); reset accumulator.
__global__ void ot_col_update(float* __restrict__ s, float* __restrict__ colsum) {
  int e = threadIdx.x;
  if (e < N_EXP) { s[e] += logf(colsum[e]); colsum[e] = 0.0f; }
}

__global__ void ot_init(float* __restrict__ s, float* __restrict__ colsum,
                        float* __restrict__ loadsum) {
  int e = threadIdx.x;
  if (e < N_EXP) { s[e] = 0.0f; colsum[e] = 0.0f; loadsum[e] = 0.0f; }
}

// ---------------------------------------------------------------------------
// Kernel 3: finalize.  pi = exp(c - r - s); write dispatch, stable top-8
// indices (tie-break lowest index, branch-free lexicographic scan, no dynamic
// register indexing), and accumulate column sums of pi for the load loss.
// ---------------------------------------------------------------------------
__global__ __launch_bounds__(256) void ot_finalize(const float* __restrict__ cost,
                                                   const float* __restrict__ s,
                                                   const float* __restrict__ r,
                                                   float* __restrict__ out,
                                                   float* __restrict__ loadsum) {
  __shared__ float s_sh[N_EXP];
  __shared__ float part[N_EXP];
  const int tid = threadIdx.x;
  if (tid < N_EXP) { s_sh[tid] = s[tid]; part[tid] = 0.0f; }
  __syncthreads();

  const long t = (long)blockIdx.x * blockDim.x + tid;
  const float rt = r[t];
  const float4* crow = (const float4*)(cost + t * N_EXP);
  float4* drow = (float4*)(out + t * N_EXP);

  float la[N_EXP];
#pragma unroll
  for (int i = 0; i < 16; ++i) {
    float4 q = crow[i];
    la[4*i+0] = q.x - rt - s_sh[4*i+0];
    la[4*i+1] = q.y - rt - s_sh[4*i+1];
    la[4*i+2] = q.z - rt - s_sh[4*i+2];
    la[4*i+3] = q.w - rt - s_sh[4*i+3];
    float4 p;
    p.x = expf(la[4*i+0]); p.y = expf(la[4*i+1]);
    p.z = expf(la[4*i+2]); p.w = expf(la[4*i+3]);
    drow[i] = p;
    atomicAdd(&part[4*i+0], p.x);
    atomicAdd(&part[4*i+1], p.y);
    atomicAdd(&part[4*i+2], p.z);
    atomicAdd(&part[4*i+3], p.w);
  }

  // top-8 without mutating la[]: next-best under (value desc, index asc).
  float* idxout = out + N_TOK * N_EXP + t * N_ACT;
  float pv = INFINITY;
  int   pix = -1;
#pragma unroll
  for (int j = 0; j < N_ACT; ++j) {
    float bv = -INFINITY;
    int   bi = 0;
#pragma unroll
    for (int e = 0; e < N_EXP; ++e) {
      bool elig   = (la[e] < pv) || ((la[e] == pv) && (e > pix));
      bool better = elig && (la[e] > bv);
      bv = better ? la[e] : bv;
      bi = better ? e     : bi;
    }
    idxout[j] = (float)bi;
    pv = bv; pix = bi;
  }

  __syncthreads();
  if (tid < N_EXP) atomicAdd(&loadsum[tid], part[tid]);
}

// load_loss = sum_e u * (log u - log(loadsum_e / N)),  u = 1/64
__global__ void ot_loss(const float* __restrict__ loadsum, float* __restrict__ out_loss) {
  __shared__ float sh[N_EXP];
  int e = threadIdx.x;
  const float u = 1.0f / (float)N_EXP;
  if (e < N_EXP) {
    float le = loadsum[e] * (1.0f / (float)N_TOK);
    sh[e] = u * (logf(u) - logf(le));
  }
  __syncthreads();
  if (e == 0) {
    float acc = 0.0f;
#pragma unroll
    for (int i = 0; i < N_EXP; ++i) acc += sh[i];
    out_loss[0] = acc;
  }
}

// ---------------------------------------------------------------------------
extern "C" void kernel_launch(void* const* d_in, const int* in_sizes, int n_in,
                              void* d_out, int out_size, void* d_ws, size_t ws_size,
                              hipStream_t stream) {
  const float* x  = (const float*)d_in[0];
  const float* gw = (const float*)d_in[1];
  float* out = (float*)d_out;
  float* ws  = (float*)d_ws;

  // Workspace layout (floats): cost (N*64) | r (N) | s (64) | colsum (64) | loadsum (64)
  float* cost    = ws;
  float* r       = ws + N_TOK * N_EXP;
  float* s       = r + N_TOK;
  float* colsum  = s + N_EXP;
  float* loadsum = colsum + N_EXP;

  // 1) cost matrix via f32 WMMA: 512 blocks x 8 waves x (16 tok, 64 exp)
  ot_gemm_cost<<<512, 256, 0, stream>>>(x, gw, cost);

  // 2) Sinkhorn: 20 x (row LSE + column LSE)
  ot_init<<<1, 64, 0, stream>>>(s, colsum, loadsum);
  for (int it = 0; it < 20; ++it) {
    ot_sinkhorn_row<<<256, 256, 0, stream>>>(cost, s, r, colsum);
    ot_col_update<<<1, 64, 0, stream>>>(s, colsum);
  }

  // 3) dispatch + top-8 indices + expert load
  ot_finalize<<<256, 256, 0, stream>>>(cost, s, r, out, loadsum);
  ot_loss<<<1, 64, 0, stream>>>(loadsum, out + N_TOK * N_EXP + N_TOK * N_ACT);
}